// DTCWTForward_77146202570926
// MI455X (gfx1250) — compile-verified
//
#include <hip/hip_runtime.h>

typedef __attribute__((ext_vector_type(2))) float v2f;
typedef __attribute__((ext_vector_type(8))) float v8f;

#if defined(__HIP_DEVICE_COMPILE__)
#if !__has_builtin(__builtin_amdgcn_wmma_f32_16x16x4_f32)
#error "missing __builtin_amdgcn_wmma_f32_16x16x4_f32 on device toolchain"
#endif
#endif

#define SQ12 0.70710678118654752440f

__device__ __forceinline__ v8f wmma_f32(v2f a, v2f b, v8f c) {
#if defined(__HIP_DEVICE_COMPILE__)
  // 8 args: (neg_a, A, neg_b, B, c_mod, C, reuse_a, reuse_b)
  return __builtin_amdgcn_wmma_f32_16x16x4_f32(false, a, false, b, (short)0, c,
                                               false, false);
#else
  c[0] += a.x * b.x;  // host-pass stub; never executed
  return c;
#endif
}

// 'symmetric' padding index (reflect including edge); halo < n always here.
// Branchless (lowers to cndmask, no exec divergence).
__device__ __forceinline__ int symref(int i, int n) {
  i = (i < 0) ? (-1 - i) : i;
  i = (i >= n) ? (2 * n - 1 - i) : i;
  return i;
}

// q2c of three 16x16 full-rate subband tiles held in LDS (s_q: [3][16][16],
// order hl, hh, lh), writing the interleaved complex yh layout
// [nc][6][Q][Q][2].  bands: z1(hl)->0, z1(hh)->1, z1(lh)->2,
//                          z2(lh)->3, z2(hh)->4, z2(hl)->5.
__device__ __forceinline__ void q2c_write(const float* s_q, float* yh, int nc,
                                          int Q, int qr0, int qc0, int lane) {
  for (int idx = lane; idx < 64; idx += 32) {
    int i = idx >> 3, j = idx & 7;
#pragma unroll
    for (int m = 0; m < 3; ++m) {
      const float* y = s_q + m * 256;
      float a = y[(2 * i) * 16 + 2 * j];
      float b = y[(2 * i) * 16 + 2 * j + 1];
      float c = y[(2 * i + 1) * 16 + 2 * j];
      float d = y[(2 * i + 1) * 16 + 2 * j + 1];
      int b1 = m, b2 = 5 - m;
      size_t p1 = ((((size_t)nc * 6 + b1) * Q + (qr0 + i)) * Q + (qc0 + j)) * 2;
      size_t p2 = ((((size_t)nc * 6 + b2) * Q + (qr0 + i)) * Q + (qc0 + j)) * 2;
      v2f z1, z2;
      z1.x = (a - d) * SQ12;
      z1.y = (b + c) * SQ12;
      z2.x = (a + d) * SQ12;
      z2.y = (b - c) * SQ12;
      *(v2f*)(yh + p1) = z1;  // 8B-aligned: p1 is even, buffers 8B-aligned
      *(v2f*)(yh + p2) = z2;
    }
  }
}

#define ZERO8 {0.f, 0.f, 0.f, 0.f, 0.f, 0.f, 0.f, 0.f}

// ---------------------------------------------------------------------------
// Level 0: colfilter by h0o/h1o then rowfilter by h0o/h1o, fused per 16x16
// tile; writes lolo (512x512/ch) to ws and yh0 (q2c'd) to d_out.
// grid: 12288 blocks x 128 thr (4 waves, 1 tile each); 48ch * 32*32 tiles.
// Band matrices: F0[i][k] = h0o[k-i-1], F1[i][k] = h1o[k-i]  (K window = 24).
// Stored as zero-padded LDS tables -> band element = 1 unconditional ds_load.
// ---------------------------------------------------------------------------
__global__ __launch_bounds__(128) void dtcwt_level0(
    const float* __restrict__ x, const float* __restrict__ g_h0o,
    const float* __restrict__ g_h1o, float* __restrict__ lolo0,
    float* __restrict__ yh0) {
  __shared__ float smem[4][2 * 16 * 48 + 3 * 16 * 16];
  __shared__ float p0[64], p1[64];  // p*[16+t] = tap t, zeros elsewhere
  if (threadIdx.x < 64) {
    int t = (int)threadIdx.x - 16;
    p0[threadIdx.x] = (t >= 0 && t < 5) ? g_h0o[t] : 0.f;
    p1[threadIdx.x] = (t >= 0 && t < 7) ? g_h1o[t] : 0.f;
  }
  __syncthreads();

  const int lane = threadIdx.x & 31;
  const int w = threadIdx.x >> 5;
  float* s_lo = smem[w];
  float* s_hi = s_lo + 16 * 48;
  float* s_q = s_hi + 16 * 48;

  int job = blockIdx.x * 4 + w;  // 48 * 1024 jobs
  int nc = job >> 10;
  int t = job & 1023;
  int r0 = (t >> 5) * 16;
  int c0 = (t & 31) * 16;
  const float* X = x + (size_t)nc * 512 * 512;

  const int al = lane & 15;
  const int kh = (lane >> 4) << 1;  // A/B lane-half K offset (0 or 2)
  const int rbase = (lane >> 4) * 8;

  // per-lane constant band table bases: F0 -> p0[15-al+kb], F1 -> p1[16-al+kb]
  const int b0base = 15 - al;
  const int b1base = 16 - al;

  // loop-invariant column base pointers for the B operand (symmetric reflect)
  const float* colp[2];
#pragma unroll
  for (int g = 0; g < 2; ++g)
    colp[g] = X + symref(c0 - 8 + g * 16 + al, 512);

  // ---- column stage: rows r0..r0+15, cols [c0-8, c0+24), K window = 24 rows
  v8f accL[2] = {ZERO8, ZERO8};
  v8f accH[2] = {ZERO8, ZERO8};
  const int wb = r0 - 3;
#pragma unroll
  for (int kc = 0; kc < 6; ++kc) {
    int kb = kc * 4 + kh;
    v2f aL, aH;
    aL.x = p0[b0base + kb];
    aL.y = p0[b0base + kb + 1];
    aH.x = p1[b1base + kb];
    aH.y = p1[b1base + kb + 1];
    size_t oA = (size_t)symref(wb + kb, 512) * 512;
    size_t oB = (size_t)symref(wb + kb + 1, 512) * 512;
#pragma unroll
    for (int g = 0; g < 2; ++g) {
      v2f b;
      b.x = colp[g][oA];
      b.y = colp[g][oB];
      accL[g] = wmma_f32(aL, b, accL[g]);
      accH[g] = wmma_f32(aH, b, accH[g]);
    }
  }
#pragma unroll
  for (int v = 0; v < 8; ++v) {
    s_lo[(rbase + v) * 48 + al] = accL[0][v];
    s_lo[(rbase + v) * 48 + 16 + al] = accL[1][v];
    s_hi[(rbase + v) * 48 + al] = accH[0][v];
    s_hi[(rbase + v) * 48 + 16 + al] = accH[1][v];
  }
  __syncthreads();

  // ---- row stage: window cols c0-3..c0+20 -> lds cols 5..28
  // G0[k][j] = h0o[k-j-1], G1[k][j] = h1o[k-j]: same tables, j = al.
  v8f ll = ZERO8, hl = ZERO8, lh = ZERO8, hh = ZERO8;
#pragma unroll
  for (int kc = 0; kc < 6; ++kc) {
    int kb = kc * 4 + kh;
    v2f aLo, aHi, b0, b1;
    aLo.x = s_lo[al * 48 + kb + 5];
    aLo.y = s_lo[al * 48 + kb + 6];
    aHi.x = s_hi[al * 48 + kb + 5];
    aHi.y = s_hi[al * 48 + kb + 6];
    b0.x = p0[b0base + kb];
    b0.y = p0[b0base + kb + 1];
    b1.x = p1[b1base + kb];
    b1.y = p1[b1base + kb + 1];
    ll = wmma_f32(aLo, b0, ll);
    hl = wmma_f32(aHi, b0, hl);
    lh = wmma_f32(aLo, b1, lh);
    hh = wmma_f32(aHi, b1, hh);
  }
#pragma unroll
  for (int v = 0; v < 8; ++v) {
    lolo0[(size_t)nc * 512 * 512 + (size_t)(r0 + rbase + v) * 512 + c0 + al] =
        ll[v];
    s_q[0 * 256 + (rbase + v) * 16 + al] = hl[v];
    s_q[1 * 256 + (rbase + v) * 16 + al] = hh[v];
    s_q[2 * 256 + (rbase + v) * 16 + al] = lh[v];
  }
  __syncthreads();
  q2c_write(s_q, yh0, nc, 256, r0 >> 1, c0 >> 1, lane);
}

// ---------------------------------------------------------------------------
// Levels 1..: qshift dual-tree stride-2 filtering, fused col+row per 16x16
// output tile (of the S/2 grid).  Writes lolo (S/2 x S/2) and yh (q2c'd).
// Band: out row i, window k (base 2*R0-8):
//   low : even i -> fa(h0b) at t2 = 2i+18-k ; odd i -> fb(h0a) at t2 = 2i+17-k
//   high: even i -> fb(h1a) at t2 = 2i+19-k ; odd i -> fa(h1b) at t2 = 2i+16-k
// value = h[t2/2] if t2 even in [0,18] else 0 -> padded table T[32+t2].
// ---------------------------------------------------------------------------
__global__ __launch_bounds__(128) void dtcwt_levelq(
    const float* __restrict__ Lin, int S, const float* __restrict__ g_h0a,
    const float* __restrict__ g_h0b, const float* __restrict__ g_h1a,
    const float* __restrict__ g_h1b, float* __restrict__ loloOut,
    float* __restrict__ yhOut) {
  __shared__ float smem[4][2 * 16 * 48 + 3 * 16 * 16];
  __shared__ float qe_lo[128], qo_lo[128], qe_hi[128], qo_hi[128];
  {
    int j = threadIdx.x;  // blockDim.x == 128
    int t2 = j - 32;
    bool vld = (t2 >= 0) && (t2 <= 18) && ((t2 & 1) == 0);
    int tt = vld ? (t2 >> 1) : 0;
    qe_lo[j] = vld ? g_h0b[tt] : 0.f;
    qo_lo[j] = vld ? g_h0a[tt] : 0.f;
    qe_hi[j] = vld ? g_h1a[tt] : 0.f;
    qo_hi[j] = vld ? g_h1b[tt] : 0.f;
  }
  __syncthreads();

  const int lane = threadIdx.x & 31;
  const int w = threadIdx.x >> 5;
  float* s_lo = smem[w];
  float* s_hi = s_lo + 16 * 48;
  float* s_q = s_hi + 16 * 48;

  const int outS = S >> 1;
  const int tpr = outS >> 4;
  int job = blockIdx.x * 4 + w;
  int nc = job / (tpr * tpr);
  int t = job % (tpr * tpr);
  int r0 = (t / tpr) * 16;
  int c0 = (t % tpr) * 16;
  const float* X = Lin + (size_t)nc * S * S;

  const int al = lane & 15;
  const int kh = (lane >> 4) << 1;
  const int rbase = (lane >> 4) * 8;
  const int W0 = 2 * r0 - 8;   // input row window base, K = 48
  const int CB = 2 * c0 - 8;   // input col window base, 48 cols

  // per-lane band table + base (index = base - kb; .y element at base - kb - 1)
  const float* Tlo = (al & 1) ? qo_lo : qe_lo;
  const float* Thi = (al & 1) ? qo_hi : qe_hi;
  const int bLo = 32 + 2 * al + ((al & 1) ? 17 : 18);
  const int bHi = 32 + 2 * al + ((al & 1) ? 16 : 19);

  // loop-invariant column base pointers
  const float* colp[3];
#pragma unroll
  for (int g = 0; g < 3; ++g)
    colp[g] = X + symref(CB + g * 16 + al, S);

  // ---- column stage: lo = dfilt(L,h0b,h0a,low), hi = dfilt(L,h1b,h1a,high)
  v8f accL[3] = {ZERO8, ZERO8, ZERO8};
  v8f accH[3] = {ZERO8, ZERO8, ZERO8};
#pragma unroll
  for (int kc = 0; kc < 12; ++kc) {
    int kb = kc * 4 + kh;
    v2f aL, aH;
    aL.x = Tlo[bLo - kb];
    aL.y = Tlo[bLo - kb - 1];
    aH.x = Thi[bHi - kb];
    aH.y = Thi[bHi - kb - 1];
    size_t oA = (size_t)symref(W0 + kb, S) * S;
    size_t oB = (size_t)symref(W0 + kb + 1, S) * S;
#pragma unroll
    for (int g = 0; g < 3; ++g) {
      v2f b;
      b.x = colp[g][oA];
      b.y = colp[g][oB];
      accL[g] = wmma_f32(aL, b, accL[g]);
      accH[g] = wmma_f32(aH, b, accH[g]);
    }
  }
#pragma unroll
  for (int v = 0; v < 8; ++v) {
#pragma unroll
    for (int g = 0; g < 3; ++g) {
      s_lo[(rbase + v) * 48 + g * 16 + al] = accL[g][v];
      s_hi[(rbase + v) * 48 + g * 16 + al] = accH[g][v];
    }
  }
  __syncthreads();

  // ---- row stage on the 48-wide LDS window; B bands use same tables (j=al)
  v8f ll = ZERO8, hl = ZERO8, lh = ZERO8, hh = ZERO8;
#pragma unroll
  for (int kc = 0; kc < 12; ++kc) {
    int kb = kc * 4 + kh;
    v2f aLo, aHi, b0, b1;
    aLo.x = s_lo[al * 48 + kb];
    aLo.y = s_lo[al * 48 + kb + 1];
    aHi.x = s_hi[al * 48 + kb];
    aHi.y = s_hi[al * 48 + kb + 1];
    b0.x = Tlo[bLo - kb];
    b0.y = Tlo[bLo - kb - 1];
    b1.x = Thi[bHi - kb];
    b1.y = Thi[bHi - kb - 1];
    ll = wmma_f32(aLo, b0, ll);
    hl = wmma_f32(aHi, b0, hl);
    lh = wmma_f32(aLo, b1, lh);
    hh = wmma_f32(aHi, b1, hh);
  }
#pragma unroll
  for (int v = 0; v < 8; ++v) {
    loloOut[(size_t)nc * outS * outS + (size_t)(r0 + rbase + v) * outS + c0 +
            al] = ll[v];
    s_q[0 * 256 + (rbase + v) * 16 + al] = hl[v];
    s_q[1 * 256 + (rbase + v) * 16 + al] = hh[v];
    s_q[2 * 256 + (rbase + v) * 16 + al] = lh[v];
  }
  __syncthreads();
  q2c_write(s_q, yhOut, nc, outS >> 1, r0 >> 1, c0 >> 1, lane);
}

extern "C" void kernel_launch(void* const* d_in, const int* in_sizes, int n_in,
                              void* d_out, int out_size, void* d_ws,
                              size_t ws_size, hipStream_t stream) {
  const float* x = (const float*)d_in[0];
  const float* h0o = (const float*)d_in[1];
  const float* h1o = (const float*)d_in[2];
  const float* h0a = (const float*)d_in[3];
  const float* h0b = (const float*)d_in[4];
  const float* h1a = (const float*)d_in[5];
  const float* h1b = (const float*)d_in[6];

  float* out = (float*)d_out;
  float* lolo_final = out;                 // (16,3,128,128)
  float* yh0 = out + 786432;               // (16,3,6,256,256,2)
  float* yh1 = out + 38535168;             // (16,3,6,128,128,2)
  float* yh2 = out + 47972352;             // (16,3,6,64,64,2)

  float* ws0 = (float*)d_ws;               // lolo after L0: 48*512*512 f32
  float* ws1 = ws0 + (size_t)48 * 512 * 512;  // lolo after L1: 48*256*256

  // level 0: 48 channels * 32*32 tiles, 4 tiles (waves) per block
  dtcwt_level0<<<12288, 128, 0, stream>>>(x, h0o, h1o, ws0, yh0);
  // level 1: 48 * 16*16 tiles
  dtcwt_levelq<<<3072, 128, 0, stream>>>(ws0, 512, h0a, h0b, h1a, h1b, ws1,
                                         yh1);
  // level 2: 48 * 8*8 tiles; final lolo goes straight to d_out
  dtcwt_levelq<<<768, 128, 0, stream>>>(ws1, 256, h0a, h0b, h1a, h1b,
                                        lolo_final, yh2);
}